// HierarchicalPooling_36266703847508
// MI455X (gfx1250) — compile-verified
//
#include <hip/hip_runtime.h>
#include <math.h>

#define N_NODES 4096
#define S_SAMP 32
#define D_MODEL_ 256
#define K_ATOMS 64
#define B_GRAPH 64
#define MAX_ITER 100
#define EPS_NODE 0.3f
#define EPS_GRAPH 0.1f
#define TINYF 1e-12f

typedef __attribute__((ext_vector_type(2))) float v2f;
typedef __attribute__((ext_vector_type(8))) float v8f;

__device__ __forceinline__ void lse_update(float& m, float& s, float t) {
    if (t > m) { s = s * __expf(m - t) + 1.0f; m = t; }
    else       { s += __expf(t - m); }
}
__device__ __forceinline__ void lse_merge(float& m, float& s, float mo, float so) {
    float M = fmaxf(m, mo);
    s = s * __expf(m - M) + so * __expf(mo - M);
    m = M;
}

// ---------------- Stage 1: fused cost(WMMA) + 100-iter Sinkhorn + node hist ----
__global__ __launch_bounds__(256) void stage1_kernel(
        const float* __restrict__ X, const float* __restrict__ CB,
        float* __restrict__ node_hists) {
    __shared__ float Klog[S_SAMP * 68];   // padded row stride 68 vs 64 banks
    __shared__ float x2s[S_SAMP];
    __shared__ float y2s[K_ATOMS];
    __shared__ float ush[S_SAMP];
    __shared__ float vsh[K_ATOMS];
    __shared__ float hsh[K_ATOMS];
    __shared__ float tot;

    const int n = blockIdx.x;
    const int tid = threadIdx.x;

    if (tid < S_SAMP)  x2s[tid] = 0.f;
    if (tid < K_ATOMS) { y2s[tid] = 0.f; vsh[tid] = 0.f; }
    __syncthreads();

    // ||x_s||^2 for the node's 32 samples
    {
        int r = tid & 31, c = tid >> 5;
        const float* p = X + (size_t)n * S_SAMP * D_MODEL_ + (size_t)r * D_MODEL_ + c * 32;
        float s = 0.f;
        #pragma unroll
        for (int d = 0; d < 32; ++d) { float v = p[d]; s += v * v; }
        atomicAdd(&x2s[r], s);
    }
    // ||codebook_k||^2
    {
        int k = tid & 63, c = tid >> 6;
        const float* p = CB + (size_t)k * D_MODEL_ + c * 64;
        float s = 0.f;
        #pragma unroll
        for (int d = 0; d < 64; ++d) { float v = p[d]; s += v * v; }
        atomicAdd(&y2s[k], s);
    }
    __syncthreads();

    // C = x2 + y2 - 2 * X[n] * CB^T  via V_WMMA_F32_16X16X4_F32
    // 8 waves -> 2 (M) x 4 (N) tiles of 16x16; 64 k-steps of K=4
    {
        int lane = tid & 31, wv = tid >> 5;
        int mr = wv & 1, nc = wv >> 1;
        int hi = lane >> 4, lm = lane & 15;
        const float* Arow = X + (size_t)n * S_SAMP * D_MODEL_ + (size_t)(mr * 16 + lm) * D_MODEL_;
        const float* Brow = CB + (size_t)(nc * 16 + lm) * D_MODEL_;
        v8f acc = {};
        #pragma unroll 8
        for (int t = 0; t < 64; ++t) {
            int k0 = 4 * t + 2 * hi;
            float2 af = *reinterpret_cast<const float2*>(Arow + k0);
            float2 bf = *reinterpret_cast<const float2*>(Brow + k0);
            v2f a; a.x = af.x; a.y = af.y;
            v2f b; b.x = bf.x; b.y = bf.y;
            acc = __builtin_amdgcn_wmma_f32_16x16x4_f32(
                false, a, false, b, (short)0, acc, false, false);
        }
        #pragma unroll
        for (int r = 0; r < 8; ++r) {
            int m  = mr * 16 + 8 * hi + r;
            int kk = nc * 16 + lm;
            float c = x2s[m] + y2s[kk] - 2.0f * acc[r];
            Klog[m * 68 + kk] = -fmaxf(c, 0.f) * (1.0f / EPS_NODE);
        }
    }
    __syncthreads();

    const float log_a = __logf(1.0f / S_SAMP + TINYF);
    const float log_b = __logf(1.0f / K_ATOMS + TINYF);

    for (int it = 0; it < MAX_ITER; ++it) {
        // u[s] = log_a - LSE_k(Klog[s][k] + v[k]); 8 lanes per row (wave32 butterfly)
        {
            int s = tid >> 3, j = tid & 7;
            float m = -1e30f, ssum = 0.f;
            #pragma unroll
            for (int i = 0; i < 8; ++i) {
                int k = j * 8 + i;
                lse_update(m, ssum, Klog[s * 68 + k] + vsh[k]);
            }
            #pragma unroll
            for (int off = 1; off < 8; off <<= 1) {
                float mo = __shfl_xor(m, off);
                float so = __shfl_xor(ssum, off);
                lse_merge(m, ssum, mo, so);
            }
            if (j == 0) ush[s] = log_a - (m + __logf(ssum));
        }
        __syncthreads();
        // v[k] = log_b - LSE_s(Klog[s][k] + u[s]); 4 lanes per column
        {
            int k = tid >> 2, j = tid & 3;
            float m = -1e30f, ssum = 0.f;
            #pragma unroll
            for (int i = 0; i < 8; ++i) {
                int s = j * 8 + i;
                lse_update(m, ssum, Klog[s * 68 + k] + ush[s]);
            }
            #pragma unroll
            for (int off = 1; off < 4; off <<= 1) {
                float mo = __shfl_xor(m, off);
                float so = __shfl_xor(ssum, off);
                lse_merge(m, ssum, mo, so);
            }
            if (j == 0) vsh[k] = log_b - (m + __logf(ssum));
        }
        __syncthreads();
    }

    // node hist: h[k] = exp(LSE_s(u + Klog) + v[k]), normalized
    {
        int k = tid >> 2, j = tid & 3;
        float m = -1e30f, ssum = 0.f;
        #pragma unroll
        for (int i = 0; i < 8; ++i) {
            int s = j * 8 + i;
            lse_update(m, ssum, Klog[s * 68 + k] + ush[s]);
        }
        #pragma unroll
        for (int off = 1; off < 4; off <<= 1) {
            float mo = __shfl_xor(m, off);
            float so = __shfl_xor(ssum, off);
            lse_merge(m, ssum, mo, so);
        }
        if (j == 0) hsh[k] = __expf(m + __logf(ssum) + vsh[k]);
    }
    __syncthreads();
    if (tid == 0) {
        float t = 0.f;
        for (int k = 0; k < K_ATOMS; ++k) t += hsh[k];
        tot = t + TINYF;
    }
    __syncthreads();
    if (tid < K_ATOMS)
        node_hists[(size_t)n * K_ATOMS + tid] = hsh[tid] / tot;
}

// ---------------- Stage 2 helpers ----------------
__global__ void atomD_kernel(const float* __restrict__ CB, float* __restrict__ aD) {
    int idx = blockIdx.x * blockDim.x + threadIdx.x;  // 64*64
    int j = idx >> 6, k = idx & 63;
    float s = 0.f;
    #pragma unroll 8
    for (int d = 0; d < D_MODEL_; ++d) {
        float df = CB[j * D_MODEL_ + d] - CB[k * D_MODEL_ + d];
        s += df * df;
    }
    aD[idx] = s;
}

__global__ void counts_kernel(const int* __restrict__ bidx, int* __restrict__ counts) {
    int b = threadIdx.x;
    if (b < B_GRAPH) {
        int c = 0;
        for (int n = 0; n < N_NODES; ++n) c += (bidx[n] == b) ? 1 : 0;
        counts[b] = c;
    }
}

__global__ void klog2_kernel(const float* __restrict__ nh, const float* __restrict__ aD,
                             float* __restrict__ Klog2) {
    int idx = blockIdx.x * blockDim.x + threadIdx.x;  // N*K
    int n = idx >> 6, k = idx & 63;
    float s = 0.f;
    #pragma unroll 8
    for (int j = 0; j < K_ATOMS; ++j)
        s += nh[n * K_ATOMS + j] * aD[j * K_ATOMS + k];
    Klog2[idx] = -s * (1.0f / EPS_GRAPH);
}

__global__ void loga2_kernel(const int* __restrict__ bidx, const int* __restrict__ counts,
                             float* __restrict__ log_a2) {
    int idx = blockIdx.x * blockDim.x + threadIdx.x;  // B*N
    int b = idx >> 12, n = idx & (N_NODES - 1);
    float a = (bidx[n] == b) ? (1.0f / fmaxf((float)counts[b], 1.0f)) : 0.0f;
    log_a2[idx] = __logf(a + TINYF);
}

__global__ void zerov2_kernel(float* __restrict__ v2) {
    int idx = blockIdx.x * blockDim.x + threadIdx.x;
    v2[idx] = 0.f;
}

// u2[b][n] = log_a2 - LSE_k(Klog2[n][k] + v2[b][k])
__global__ __launch_bounds__(256) void u2_kernel(const float* __restrict__ Klog2,
        const float* __restrict__ log_a2, const float* __restrict__ v2,
        float* __restrict__ u2) {
    __shared__ float vsh[K_ATOMS];
    int b = blockIdx.y;
    int tid = threadIdx.x;
    if (tid < K_ATOMS) vsh[tid] = v2[b * K_ATOMS + tid];
    __syncthreads();
    int n = blockIdx.x * blockDim.x + tid;
    const float4* row = reinterpret_cast<const float4*>(Klog2 + (size_t)n * K_ATOMS);
    float m = -1e30f, s = 0.f;
    #pragma unroll 4
    for (int q = 0; q < 16; ++q) {
        float4 kv = row[q];
        lse_update(m, s, kv.x + vsh[4 * q + 0]);
        lse_update(m, s, kv.y + vsh[4 * q + 1]);
        lse_update(m, s, kv.z + vsh[4 * q + 2]);
        lse_update(m, s, kv.w + vsh[4 * q + 3]);
    }
    size_t o = (size_t)b * N_NODES + n;
    u2[o] = log_a2[o] - (m + __logf(s));
}

// partial column LSEs over a 512-row chunk; 8 chunks x 64 graphs
__global__ __launch_bounds__(256) void v2_partial_kernel(const float* __restrict__ Klog2,
        const float* __restrict__ u2, float* __restrict__ part) {
    __shared__ float mArr[256], sArr[256];
    int chunk = blockIdx.x;   // 0..7
    int b = blockIdx.y;
    int tid = threadIdx.x;
    int k = tid & 63, p = tid >> 2 >> 4;  // p = tid/64
    p = tid >> 6;
    int nbase = chunk * 512;
    float m = -1e30f, s = 0.f;
    for (int i = 0; i < 128; ++i) {
        int n = nbase + p + 4 * i;
        lse_update(m, s, Klog2[(size_t)n * K_ATOMS + k] + u2[(size_t)b * N_NODES + n]);
    }
    mArr[tid] = m; sArr[tid] = s;
    __syncthreads();
    if (tid < 64) {
        for (int q = 1; q < 4; ++q) lse_merge(m, s, mArr[tid + 64 * q], sArr[tid + 64 * q]);
        int o = (b * 8 + chunk) * 64 + tid;
        part[2 * o] = m; part[2 * o + 1] = s;
    }
}

__global__ void v2_combine_kernel(const float* __restrict__ part, float* __restrict__ v2) {
    int idx = blockIdx.x * blockDim.x + threadIdx.x;  // B*K
    int b = idx >> 6, k = idx & 63;
    float m = -1e30f, s = 0.f;
    #pragma unroll
    for (int c = 0; c < 8; ++c) {
        int o = (b * 8 + c) * 64 + k;
        lse_merge(m, s, part[2 * o], part[2 * o + 1]);
    }
    const float log_b = __logf(1.0f / K_ATOMS + TINYF);
    v2[idx] = log_b - (m + __logf(s));
}

__global__ __launch_bounds__(256) void final_kernel(const float* __restrict__ Klog2,
        const float* __restrict__ u2, const float* __restrict__ v2,
        const int* __restrict__ counts, float* __restrict__ out) {
    __shared__ float mArr[256], sArr[256];
    __shared__ float hsh[K_ATOMS];
    __shared__ float tot;
    int b = blockIdx.x;
    int tid = threadIdx.x;
    int k = tid & 63, p = tid >> 6;
    float m = -1e30f, s = 0.f;
    for (int i = 0; i < N_NODES / 4; ++i) {
        int n = p + 4 * i;
        lse_update(m, s, Klog2[(size_t)n * K_ATOMS + k] + u2[(size_t)b * N_NODES + n]);
    }
    mArr[tid] = m; sArr[tid] = s;
    __syncthreads();
    if (tid < 64) {
        for (int q = 1; q < 4; ++q) lse_merge(m, s, mArr[tid + 64 * q], sArr[tid + 64 * q]);
        hsh[tid] = __expf(m + __logf(s) + v2[b * K_ATOMS + tid]);
    }
    __syncthreads();
    if (tid == 0) {
        float t = 0.f;
        for (int kk = 0; kk < K_ATOMS; ++kk) t += hsh[kk];
        tot = t + TINYF;
    }
    __syncthreads();
    if (tid < 64)
        out[b * K_ATOMS + tid] = (counts[b] > 0) ? (hsh[tid] / tot) : (1.0f / K_ATOMS);
}

extern "C" void kernel_launch(void* const* d_in, const int* in_sizes, int n_in,
                              void* d_out, int out_size, void* d_ws, size_t ws_size,
                              hipStream_t stream) {
    (void)in_sizes; (void)n_in; (void)out_size; (void)ws_size;
    const float* X   = (const float*)d_in[0];
    const int* bidx  = (const int*)d_in[1];
    const float* CB  = (const float*)d_in[2];
    float* out = (float*)d_out;

    char* ws = (char*)d_ws;
    size_t off = 0;
    auto alloc = [&](size_t bytes) {
        void* p = ws + off;
        off = (off + bytes + 255) & ~(size_t)255;
        return p;
    };
    float* node_hists = (float*)alloc((size_t)N_NODES * K_ATOMS * 4);
    float* aD         = (float*)alloc((size_t)K_ATOMS * K_ATOMS * 4);
    float* Klog2      = (float*)alloc((size_t)N_NODES * K_ATOMS * 4);
    float* log_a2     = (float*)alloc((size_t)B_GRAPH * N_NODES * 4);
    float* u2         = (float*)alloc((size_t)B_GRAPH * N_NODES * 4);
    float* v2         = (float*)alloc((size_t)B_GRAPH * K_ATOMS * 4);
    float* part       = (float*)alloc((size_t)B_GRAPH * 8 * K_ATOMS * 2 * 4);
    int*   counts     = (int*)alloc((size_t)B_GRAPH * 4);

    stage1_kernel<<<N_NODES, 256, 0, stream>>>(X, CB, node_hists);
    atomD_kernel<<<(K_ATOMS * K_ATOMS) / 256, 256, 0, stream>>>(CB, aD);
    counts_kernel<<<1, 64, 0, stream>>>(bidx, counts);
    klog2_kernel<<<(N_NODES * K_ATOMS) / 256, 256, 0, stream>>>(node_hists, aD, Klog2);
    loga2_kernel<<<(B_GRAPH * N_NODES) / 256, 256, 0, stream>>>(bidx, counts, log_a2);
    zerov2_kernel<<<(B_GRAPH * K_ATOMS) / 256, 256, 0, stream>>>(v2);
    for (int it = 0; it < MAX_ITER; ++it) {
        u2_kernel<<<dim3(N_NODES / 256, B_GRAPH), 256, 0, stream>>>(Klog2, log_a2, v2, u2);
        v2_partial_kernel<<<dim3(8, B_GRAPH), 256, 0, stream>>>(Klog2, u2, part);
        v2_combine_kernel<<<(B_GRAPH * K_ATOMS) / 256, 256, 0, stream>>>(part, v2);
    }
    final_kernel<<<B_GRAPH, 256, 0, stream>>>(Klog2, u2, v2, counts, out);
}